// GNN_20194936226696
// MI455X (gfx1250) — compile-verified
//
#include <hip/hip_runtime.h>
#include <cstddef>
#include <cstdint>

#define N_NODES  100000
#define N_EDGES  1600000
#define IN_CH    128
#define HID      128
#define OUT_CH   16
#define N_GRAPHS 64

typedef __attribute__((ext_vector_type(2))) float v2f;
typedef __attribute__((ext_vector_type(8))) float v8f;

// ---------------------------------------------------------------------------
// init: deg=1 (self loop), cnt=0, fill=0
// ---------------------------------------------------------------------------
__global__ void k_init(float* deg, int* cnt, int* fill) {
    int i = blockIdx.x * blockDim.x + threadIdx.x;
    if (i < N_NODES) { deg[i] = 1.0f; cnt[i] = 0; fill[i] = 0; }
}

// per-edge degree count (int-valued float adds -> deterministic)
__global__ void k_count(const long long* __restrict__ src,
                        const long long* __restrict__ dst,
                        float* deg, int* cnt) {
    int e = blockIdx.x * blockDim.x + threadIdx.x;
    if (e >= N_EDGES) return;
    int d = (int)dst[e];
    atomicAdd(&deg[d], 1.0f);
    atomicAdd(&cnt[d], 1);
}

__global__ void k_degfin(const float* __restrict__ deg,
                         float* __restrict__ dis, float* __restrict__ dinv) {
    int i = blockIdx.x * blockDim.x + threadIdx.x;
    if (i < N_NODES) { float d = deg[i]; dis[i] = rsqrtf(d); dinv[i] = 1.0f / d; }
}

// single-block exclusive scan (100k elements, chunked Hillis-Steele)
__global__ __launch_bounds__(1024) void k_scan(const int* __restrict__ cnt,
                                               int* __restrict__ rowptr, int N) {
    __shared__ int s[1024];
    __shared__ int carry_s;
    int tid = threadIdx.x;
    if (tid == 0) carry_s = 0;
    __syncthreads();
    for (int base = 0; base < N; base += 1024) {
        int idx = base + tid;
        int v = (idx < N) ? cnt[idx] : 0;
        s[tid] = v;
        __syncthreads();
        for (int off = 1; off < 1024; off <<= 1) {
            int add = (tid >= off) ? s[tid - off] : 0;
            __syncthreads();
            s[tid] += add;
            __syncthreads();
        }
        int incl = s[tid];
        if (idx < N) rowptr[idx] = carry_s + incl - v;
        __syncthreads();
        if (tid == 1023) carry_s += incl;
        __syncthreads();
    }
    if (tid == 0) rowptr[N] = carry_s;
}

// scatter edges into CSR; precompute edge weight dis[src]*dis[dst]
__global__ void k_fill(const long long* __restrict__ src,
                       const long long* __restrict__ dst,
                       const int* __restrict__ rowptr, int* fill,
                       const float* __restrict__ dis,
                       int* __restrict__ col, float* __restrict__ wgt) {
    int e = blockIdx.x * blockDim.x + threadIdx.x;
    if (e >= N_EDGES) return;
    int s = (int)src[e], d = (int)dst[e];
    int p = rowptr[d] + atomicAdd(&fill[d], 1);
    col[p] = s;
    wgt[p] = dis[s] * dis[d];
}

// ---------------------------------------------------------------------------
// fp32 GEMM: out[M,128] = A[M,128] @ W[128,128], exact fp32 via
// V_WMMA_F32_16X16X4_F32. 8 waves/block: wave w owns N-tile [16w,16w+16).
// W staged in LDS (64KB). X tiles (16x128, stride-132 padded) are
// DOUBLE-BUFFERED via GLOBAL_LOAD_ASYNC_TO_LDS_B128 (ASYNCcnt-tracked),
// overlapping the next tile's global->LDS DMA with the current tile's WMMAs.
// ---------------------------------------------------------------------------
__global__ __launch_bounds__(256) void k_gemm(const float* __restrict__ A,
                                              const float* __restrict__ W,
                                              float* __restrict__ out, int M) {
    extern __shared__ float lds[];
    float* Wlds = lds;                         // 128*128
    float* Xbuf0 = lds + 128 * 128;            // 16*132 (padded)
    float* Xbuf1 = Xbuf0 + 16 * 132;           // 16*132 (padded)
    const int tid  = threadIdx.x;
    const int lane = tid & 31;
    const int wave = tid >> 5;        // 0..7
    const int half = lane >> 4;       // 0/1  (K-pair select)
    const int mrow = lane & 15;       // A row within tile
    // stage W (16384 floats) via float4
    {
        const float4* Wv = (const float4*)W;
        float4* Lv = (float4*)Wlds;
        for (int i = tid; i < 4096; i += 256) Lv[i] = Wv[i];
    }
    // async-stage one 16x128 X tile (512 x B128 chunks, 2 per thread)
    auto issue_tile = [&](int t, float* Xdst) {
        const float* gsrc = A + (size_t)t * 2048;
        for (int c = tid; c < 512; c += 256) {
            int r = c >> 5, cv = c & 31;
            unsigned loff = (unsigned)(uintptr_t)(&Xdst[r * 132 + cv * 4]);
            const float* g = gsrc + c * 4;
            asm volatile("global_load_async_to_lds_b128 %0, %1, off"
                         :: "v"(loff), "v"(g) : "memory");
        }
    };
    const int ntiles = M >> 4;
    const int ncol = wave * 16 + (lane & 15);
    int parity = 0;
    if (blockIdx.x < ntiles) issue_tile(blockIdx.x, Xbuf0);   // prologue
    for (int t = blockIdx.x; t < ntiles; t += gridDim.x) {
        // this wave's async writes into the current buffer have landed:
        asm volatile("s_wait_asynccnt 0x0" ::: "memory");
        // everyone's writes visible + everyone done reading the other buffer:
        __syncthreads();
        int tn = t + gridDim.x;
        if (tn < ntiles) issue_tile(tn, parity ? Xbuf0 : Xbuf1);
        const float* X = parity ? Xbuf1 : Xbuf0;
        v8f c = {0.f, 0.f, 0.f, 0.f, 0.f, 0.f, 0.f, 0.f};
        #pragma unroll 8
        for (int k0 = 0; k0 < 128; k0 += 4) {
            int kk = k0 + 2 * half;
            v2f a, b;
            a.x = X[mrow * 132 + kk];
            a.y = X[mrow * 132 + kk + 1];
            b.x = Wlds[kk * 128 + ncol];
            b.y = Wlds[(kk + 1) * 128 + ncol];
            c = __builtin_amdgcn_wmma_f32_16x16x4_f32(false, a, false, b,
                                                      (short)0, c, false, false);
        }
        float* orow = out + (size_t)t * 16 * 128 + ncol;
        #pragma unroll
        for (int r = 0; r < 8; ++r)
            orow[(size_t)(r + 8 * half) * 128] = c[r];
        parity ^= 1;
    }
}

// ---------------------------------------------------------------------------
// CSR aggregation: one wave per node, lane holds float4 of 4 channels.
// out = relu(sum_e w[e]*h[col[e]] + dinv[i]*h[i] + bias)
// ---------------------------------------------------------------------------
__global__ __launch_bounds__(256) void k_agg(const float* __restrict__ h,
                                             const float* __restrict__ bias,
                                             const float* __restrict__ dinv,
                                             const int* __restrict__ rowptr,
                                             const int* __restrict__ col,
                                             const float* __restrict__ wgt,
                                             float* __restrict__ out) {
    int wid  = (blockIdx.x * blockDim.x + threadIdx.x) >> 5;
    int lane = threadIdx.x & 31;
    if (wid >= N_NODES) return;
    const float4* h4 = (const float4*)h;
    float4 acc = make_float4(0.f, 0.f, 0.f, 0.f);
    int e0 = rowptr[wid], e1 = rowptr[wid + 1];
    #pragma unroll 2
    for (int e = e0; e < e1; ++e) {
        int s = col[e];
        float w = wgt[e];
        float4 v = h4[(size_t)s * 32 + lane];
        acc.x += w * v.x; acc.y += w * v.y; acc.z += w * v.z; acc.w += w * v.w;
    }
    float di = dinv[wid];
    float4 hv = h4[(size_t)wid * 32 + lane];
    float4 bv = ((const float4*)bias)[lane];
    acc.x = fmaxf(acc.x + di * hv.x + bv.x, 0.f);
    acc.y = fmaxf(acc.y + di * hv.y + bv.y, 0.f);
    acc.z = fmaxf(acc.z + di * hv.z + bv.z, 0.f);
    acc.w = fmaxf(acc.w + di * hv.w + bv.w, 0.f);
    ((float4*)out)[(size_t)wid * 32 + lane] = acc;
}

// graph boundary offsets via binary search in sorted batch
__global__ void k_offsets(const long long* __restrict__ batch, int* gstart) {
    int g = blockIdx.x * blockDim.x + threadIdx.x;
    if (g > N_GRAPHS) return;
    if (g == N_GRAPHS) { gstart[N_GRAPHS] = N_NODES; return; }
    int lo = 0, hi = N_NODES;
    while (lo < hi) { int mid = (lo + hi) >> 1; if (batch[mid] < (long long)g) lo = mid + 1; else hi = mid; }
    gstart[g] = lo;
}

// deterministic per-graph mean pool: one block per graph, one thread/channel
__global__ __launch_bounds__(128) void k_pool(const float* __restrict__ h,
                                              const int* __restrict__ gstart,
                                              float* __restrict__ pooled) {
    int g = blockIdx.x, c = threadIdx.x;
    int s = gstart[g], e = gstart[g + 1];
    float acc = 0.f;
    for (int i = s; i < e; ++i) acc += h[(size_t)i * 128 + c];
    pooled[g * 128 + c] = acc / fmaxf((float)(e - s), 1.0f);
}

// head: out[64,16] = pooled @ Wfc + bfc
__global__ void k_final(const float* __restrict__ pooled,
                        const float* __restrict__ Wfc,
                        const float* __restrict__ bfc, float* __restrict__ out) {
    int i = blockIdx.x * blockDim.x + threadIdx.x;
    if (i >= N_GRAPHS * OUT_CH) return;
    int g = i >> 4, o = i & 15;
    float acc = bfc[o];
    #pragma unroll 4
    for (int c = 0; c < 128; ++c) acc += pooled[g * 128 + c] * Wfc[c * 16 + o];
    out[i] = acc;
}

// ---------------------------------------------------------------------------
extern "C" void kernel_launch(void* const* d_in, const int* in_sizes, int n_in,
                              void* d_out, int out_size, void* d_ws, size_t ws_size,
                              hipStream_t stream) {
    const float* x   = (const float*)d_in[0];
    const float* W1  = (const float*)d_in[1];
    const float* b1  = (const float*)d_in[2];
    const float* W2  = (const float*)d_in[3];
    const float* b2  = (const float*)d_in[4];
    const float* Wfc = (const float*)d_in[5];
    const float* bfc = (const float*)d_in[6];
    const long long* edge  = (const long long*)d_in[7];
    const long long* batch = (const long long*)d_in[8];
    const long long* srcI = edge;
    const long long* dstI = edge + N_EDGES;

    char* base = (char*)d_ws;
    size_t off = 0;
    auto alloc = [&](size_t bytes) -> void* {
        void* p = base + off;
        off = (off + bytes + 511) & ~(size_t)511;
        return p;
    };
    float* deg    = (float*)alloc((size_t)N_NODES * 4);
    float* dis    = (float*)alloc((size_t)N_NODES * 4);
    float* dinv   = (float*)alloc((size_t)N_NODES * 4);
    int*   cnt    = (int*)  alloc((size_t)N_NODES * 4);
    int*   rowptr = (int*)  alloc((size_t)(N_NODES + 1) * 4);
    int*   fill   = (int*)  alloc((size_t)N_NODES * 4);
    int*   col    = (int*)  alloc((size_t)N_EDGES * 4);
    float* wgt    = (float*)alloc((size_t)N_EDGES * 4);
    int*   gstart = (int*)  alloc((size_t)(N_GRAPHS + 1) * 4);
    float* pooled = (float*)alloc((size_t)N_GRAPHS * HID * 4);
    float* h      = (float*)alloc((size_t)N_NODES * HID * 4);
    float* h2     = (float*)alloc((size_t)N_NODES * HID * 4);

    const size_t shmem = (size_t)(128 * 128 + 2 * 16 * 132) * sizeof(float);

    k_init  <<<(N_NODES + 255) / 256, 256, 0, stream>>>(deg, cnt, fill);
    k_count <<<(N_EDGES + 255) / 256, 256, 0, stream>>>(srcI, dstI, deg, cnt);
    k_degfin<<<(N_NODES + 255) / 256, 256, 0, stream>>>(deg, dis, dinv);
    k_scan  <<<1, 1024, 0, stream>>>(cnt, rowptr, N_NODES);
    k_fill  <<<(N_EDGES + 255) / 256, 256, 0, stream>>>(srcI, dstI, rowptr, fill, dis, col, wgt);

    // layer 1
    k_gemm<<<2048, 256, shmem, stream>>>(x, W1, h, N_NODES);
    k_agg <<<(N_NODES + 7) / 8, 256, 0, stream>>>(h, b1, dinv, rowptr, col, wgt, h2);
    // layer 2 (reuse h as gemm output, h2 as agg output)
    k_gemm<<<2048, 256, shmem, stream>>>(h2, W2, h, N_NODES);
    k_agg <<<(N_NODES + 7) / 8, 256, 0, stream>>>(h, b2, dinv, rowptr, col, wgt, h2);

    k_offsets<<<1, 128, 0, stream>>>(batch, gstart);
    k_pool   <<<N_GRAPHS, 128, 0, stream>>>(h2, gstart, pooled);
    k_final  <<<4, 256, 0, stream>>>(pooled, Wfc, bfc, (float*)d_out);
}